// IrrepsIndexedLinear_21672404975706
// MI455X (gfx1250) — compile-verified
//
#include <hip/hip_runtime.h>
#include <stdint.h>

typedef __attribute__((ext_vector_type(2))) float v2f;
typedef __attribute__((ext_vector_type(8))) float v8f;
typedef __attribute__((ext_vector_type(4))) unsigned int u32x4;
typedef __attribute__((ext_vector_type(4))) int          i32x4;
typedef __attribute__((ext_vector_type(8))) int          i32x8;

#define NUM_SEG 64
#define W_DIM   21504      // 128*128 + 64*64 + 32*32
#define N_ELEM  8192

// ---------------------------------------------------------------------------
// Tiny prefix-sum kernel: counts[64] -> offs[65] (exclusive scan) in d_ws.
// ---------------------------------------------------------------------------
__global__ void seg_prefix_kernel(const int* __restrict__ counts,
                                  int* __restrict__ offs) {
  if (threadIdx.x == 0) {
    int s = 0;
    for (int g = 0; g < NUM_SEG; ++g) { offs[g] = s; s += counts[g]; }
    offs[NUM_SEG] = s;
  }
}

// ---------------------------------------------------------------------------
// One 16-row M-tile of one irrep: cache A fragments in VGPRs, sweep all
// 16-wide N-tiles with B served from LDS, accumulate via f32 WMMA.
//   A: 16x4 f32  -> v2f per lane (lanes 0-15: K=k0,k0+1; lanes 16-31: K=k0+2,3)
//   B: 4x16 f32  -> v2f per lane (same K split; N = lane&15)
//   C/D: 16x16 f32 -> v8f (lanes 0-15: M=v; lanes 16-31: M=8+v; N = lane&15)
// ---------------------------------------------------------------------------
template<int MUL, int IRDIM>
__device__ __forceinline__ void irrep_mtile(
    const float* __restrict__ x, float* __restrict__ y,
    const float* __restrict__ wblk, float coeff,
    int seg_start, int rows, int mt, int lane)
{
  const int Mtot  = rows * IRDIM;
  const int num_n = MUL >> 4;
  const int mloc  = lane & 15;
  const int kh    = (lane >> 4) << 1;          // 0 or 2

  // ---- load & cache A fragments for this M-tile (x read exactly once) ----
  const int  mrow   = (mt << 4) + mloc;
  const bool mvalid = mrow < Mtot;
  const int  msafe  = mvalid ? mrow : 0;
  const int  nidx   = msafe / IRDIM;           // row within segment
  const int  comp   = msafe - nidx * IRDIM;    // irrep component i
  const float* __restrict__ xrow =
      x + (size_t)(seg_start + nidx) * (MUL * IRDIM) + comp;

  v2f afrag[MUL / 4];
#pragma unroll
  for (int k0 = 0; k0 < MUL; k0 += 4) {
    float a0 = xrow[(size_t)(k0 + kh)     * IRDIM];
    float a1 = xrow[(size_t)(k0 + kh + 1) * IRDIM];
    afrag[k0 >> 2].x = mvalid ? a0 : 0.0f;
    afrag[k0 >> 2].y = mvalid ? a1 : 0.0f;
  }

  const bool full_tile = ((mt + 1) << 4) <= Mtot;   // wave-uniform
  const int  mbase     = (mt << 4) + ((lane >> 4) << 3);

  // ---- sweep N-tiles; B from LDS ----
#pragma unroll 1
  for (int nt = 0; nt < num_n; ++nt) {
    const int ncol = (nt << 4) + mloc;
    v8f acc = {};
#pragma unroll
    for (int k0 = 0; k0 < MUL; k0 += 4) {
      v2f b;
      b.x = wblk[(k0 + kh)     * MUL + ncol];
      b.y = wblk[(k0 + kh + 1) * MUL + ncol];
      acc = __builtin_amdgcn_wmma_f32_16x16x4_f32(
          /*neg_a=*/false, afrag[k0 >> 2], /*neg_b=*/false, b,
          /*c_mod=*/(short)0, acc, /*reuse_a=*/false, /*reuse_b=*/false);
    }
    // ---- store D tile: branch-free fast path for full tiles ----
    if (full_tile) {
#pragma unroll
      for (int v = 0; v < 8; ++v) {
        const int Mr = mbase + v;
        const int n  = Mr / IRDIM;
        const int ic = Mr - n * IRDIM;
        y[((size_t)(seg_start + n) * MUL + ncol) * IRDIM + ic] = coeff * acc[v];
      }
    } else {
#pragma unroll
      for (int v = 0; v < 8; ++v) {
        const int Mr = mbase + v;
        if (Mr < Mtot) {
          const int n  = Mr / IRDIM;
          const int ic = Mr - n * IRDIM;
          y[((size_t)(seg_start + n) * MUL + ncol) * IRDIM + ic] = coeff * acc[v];
        }
      }
    }
  }
}

// ---------------------------------------------------------------------------
// Main kernel: block = (segment g, slice). Weights for g staged into LDS via
// the Tensor Data Mover (one flat 86016-byte tile), fallback to float4 loop.
// ---------------------------------------------------------------------------
__global__ __launch_bounds__(256)
void IrrepsIndexedLinear_21672404975706_kernel(
    const float* __restrict__ x0, const float* __restrict__ x1,
    const float* __restrict__ x2, const float* __restrict__ w,
    const int* __restrict__ offs,
    float* __restrict__ y0, float* __restrict__ y1, float* __restrict__ y2)
{
  __shared__ float4 lds_w4[W_DIM / 4];         // 84 KB of the 320 KB WGP LDS
  const float* lds_w = (const float*)lds_w4;

  const int g         = blockIdx.x;
  const int seg_start = offs[g];
  const int rows      = offs[g + 1] - seg_start;

#if __has_builtin(__builtin_amdgcn_tensor_load_to_lds) && \
    __has_builtin(__builtin_amdgcn_s_wait_tensorcnt)
  // ---- TDM weight stage: one 1-D tile of 10752 x 8B = 86016 bytes ----
  if ((threadIdx.x >> 5) == 0) {               // wave 0 only (uniform branch)
    const uint64_t gaddr = (uint64_t)(uintptr_t)(w + (size_t)g * W_DIM);
    const uint32_t laddr = (uint32_t)(uintptr_t)(void*)lds_w4; // low32 = LDS addr

    u32x4 g0;
    g0.x = 1u;                                  // count=1, is_restore=0, no gather
    g0.y = laddr;                               // lds_addr
    g0.z = (uint32_t)(gaddr & 0xFFFFFFFFull);   // global_addr[31:0]
    g0.w = (uint32_t)((gaddr >> 32) & 0x01FFFFFFull) | (2u << 30); // ga[56:32] | type=2

    const uint32_t nelem = (W_DIM * 4) / 8;     // 10752 8-byte elements
    i32x8 g1;
    g1[0] = (int)(3u << 16);                    // wg_mask=0, data_size=3 (8B), no flags
    g1[1] = (int)((nelem & 0xFFFFu) << 16);     // tensor_dim0[15:0]
    g1[2] = (int)((nelem >> 16) | (1u << 16));  // tensor_dim0[31:16] | tensor_dim1=1
    g1[3] = (int)((nelem & 0xFFFFu) << 16);     // tensor_dim1[31:16]=0 | tile_dim0
    g1[4] = 0;                                  // tile_dim1=0 (1-D), tile_dim2=0
    g1[5] = (int)nelem;                         // tensor_dim0_stride[31:0]
    g1[6] = 0;                                  // stride hi | dim1_stride lo
    g1[7] = 0;

    i32x4 zero4 = {0, 0, 0, 0};
#if defined(__clang_major__) && (__clang_major__ >= 23)
    i32x8 zero8 = {0, 0, 0, 0, 0, 0, 0, 0};
    __builtin_amdgcn_tensor_load_to_lds(g0, g1, zero4, zero4, zero8, 0);
#else
    __builtin_amdgcn_tensor_load_to_lds(g0, g1, zero4, zero4, 0);
#endif
    __builtin_amdgcn_s_wait_tensorcnt(0);
  }
#else
  // ---- fallback: cooperative float4 stage ----
  {
    const float4* src = (const float4*)(w + (size_t)g * W_DIM);
    for (int i = threadIdx.x; i < W_DIM / 4; i += blockDim.x)
      lds_w4[i] = src[i];
  }
#endif
  __syncthreads();
  if (rows == 0) return;                        // block-uniform exit

  const int lane = threadIdx.x & 31;
  const int wave = threadIdx.x >> 5;
  const int nvw  = gridDim.y * (blockDim.x >> 5);
  const int vw   = blockIdx.y * (blockDim.x >> 5) + wave;

  const int nm0 = (rows * 1 + 15) >> 4;
  const int nm1 = (rows * 3 + 15) >> 4;
  const int nm2 = (rows * 5 + 15) >> 4;
  const int ntask = nm0 + nm1 + nm2;

  // coeff = scale / sqrt(num_indices) / sqrt(mul_in)
  const float c0 = 0.01104854346f;              // 1/(8*sqrt(128))
  const float c1 = 0.015625f;                   // 1/(8*8)
  const float c2 = 0.02209708691f;              // 1/(8*sqrt(32))

  for (int t = vw; t < ntask; t += nvw) {       // t is wave-uniform
    if (t < nm0) {
      irrep_mtile<128, 1>(x0, y0, lds_w, c0, seg_start, rows, t, lane);
    } else if (t < nm0 + nm1) {
      irrep_mtile<64, 3>(x1, y1, lds_w + 128 * 128, c1,
                         seg_start, rows, t - nm0, lane);
    } else {
      irrep_mtile<32, 5>(x2, y2, lds_w + 128 * 128 + 64 * 64, c2,
                         seg_start, rows, t - nm0 - nm1, lane);
    }
  }
}

// ---------------------------------------------------------------------------
extern "C" void kernel_launch(void* const* d_in, const int* in_sizes, int n_in,
                              void* d_out, int out_size, void* d_ws, size_t ws_size,
                              hipStream_t stream) {
  (void)in_sizes; (void)n_in; (void)out_size; (void)ws_size;
  const float* x0     = (const float*)d_in[0];
  const float* x1     = (const float*)d_in[1];
  const float* x2     = (const float*)d_in[2];
  const float* w      = (const float*)d_in[3];
  const int*   counts = (const int*)d_in[4];

  float* y0 = (float*)d_out;
  float* y1 = y0 + (size_t)N_ELEM * 128;        // after x0-shaped output
  float* y2 = y1 + (size_t)N_ELEM * 64 * 3;     // after x1-shaped output

  int* offs = (int*)d_ws;                       // 65 ints of scratch

  seg_prefix_kernel<<<1, 64, 0, stream>>>(counts, offs);

  dim3 grid(NUM_SEG, 8);                        // 512 blocks, 8 waves each
  IrrepsIndexedLinear_21672404975706_kernel<<<grid, 256, 0, stream>>>(
      x0, x1, x2, w, offs, y0, y1, y2);
}